// AutoAdaptiveFocalLossV2_22574348108430
// MI455X (gfx1250) — compile-verified
//
#include <hip/hip_runtime.h>
#include <hip/hip_bf16.h>

// AutoAdaptiveFocalLossV2 on MI455X (gfx1250).
// Memory-bound row softmax: stream 524MB once via async-to-LDS double buffering,
// two-pass (max, exp-sum) reduction from LDS, one wave per row (wave32).

#define C_CLASSES     1000
#define ROW_BYTES     4000      // C_CLASSES * 4
#define WAVES_PER_BLK 8
#define BLOCKS        2048
#define NUM_EDGES     14        // NUM_BINS - 1
#define EPSF          1e-20f

// Issue 8 async b128 copies: 32 lanes * 16B = 512B per instr, 8 instrs = 4096B,
// with the last instr clamped to byte offset 3984 (row is 4000B; dup lanes
// rewrite identical bytes at the same LDS address -> benign).
__device__ __forceinline__ void async_row_load(unsigned lds_byte_base,
                                               const float* __restrict__ rowptr,
                                               unsigned lane) {
  unsigned long long gbase = (unsigned long long)rowptr;
#pragma unroll
  for (int i = 0; i < 8; ++i) {
    unsigned off = (unsigned)i * 512u + lane * 16u;
    if (off > 3984u) off = 3984u;            // only clamps in iteration 7
    unsigned lds_addr = lds_byte_base + off;
    unsigned long long gaddr = gbase + (unsigned long long)off;
    asm volatile("global_load_async_to_lds_b128 %0, %1, off"
                 :
                 : "v"(lds_addr), "v"(gaddr)
                 : "memory");
  }
}

__global__ __launch_bounds__(256) void focal_main_kernel(
    const float* __restrict__ logits,
    const int*   __restrict__ target,
    const float* __restrict__ bin_edges,
    const float* __restrict__ bin_gammas,
    float*       __restrict__ partials,
    int nrows)
{
  // 8 waves * 2 buffers * 1000 floats = 16000 floats (64000 B) + 8 wave accs
  __shared__ float smem[16000 + WAVES_PER_BLK];

  const int tid   = threadIdx.x;
  const int lane  = tid & 31;
  const int warp  = tid >> 5;
  const int gwave = blockIdx.x * WAVES_PER_BLK + warp;
  const int wstride = gridDim.x * WAVES_PER_BLK;

  // LDS byte address of this wave's buffer region (flat low 32 bits == LDS offset)
  const unsigned lds0 =
      (unsigned)(unsigned long long)(&smem[0]) + (unsigned)(warp * 2 * ROW_BYTES);
  const float* lbase = smem + warp * 2000;   // float view of same region

  float acc = 0.0f;
  int row = gwave;
  int buf = 0;

  if (row < nrows)
    async_row_load(lds0, logits + (long long)row * C_CLASSES, (unsigned)lane);

  for (; row < nrows; row += wstride) {
    const int nxt = row + wstride;
    if (nxt < nrows) {
      // prefetch next row into the other buffer, then wait for current row's 8 ops
      async_row_load(lds0 + (unsigned)((buf ^ 1) * ROW_BYTES),
                     logits + (long long)nxt * C_CLASSES, (unsigned)lane);
      asm volatile("s_wait_asynccnt 0x8" ::: "memory");
    } else {
      asm volatile("s_wait_asynccnt 0x0" ::: "memory");
    }

    const float* lrow = lbase + buf * 1000;

    // ---- pass 1: row max (no transcendentals) ----
    float m = -3.402823466e+38f;
#pragma unroll
    for (int i = 0; i < 7; ++i) {
      float4 v = *(const float4*)(lrow + i * 128 + lane * 4);
      m = fmaxf(m, fmaxf(fmaxf(v.x, v.y), fmaxf(v.z, v.w)));
    }
    if (lane < 26) {  // elements 896..999 = 26 float4s
      float4 v = *(const float4*)(lrow + 896 + lane * 4);
      m = fmaxf(m, fmaxf(fmaxf(v.x, v.y), fmaxf(v.z, v.w)));
    }
#pragma unroll
    for (int o = 16; o > 0; o >>= 1)
      m = fmaxf(m, __shfl_xor(m, o, 32));

    // ---- pass 2: sum exp(x - m), exactly one v_exp_f32 per element ----
    float s = 0.0f;
#pragma unroll
    for (int i = 0; i < 7; ++i) {
      float4 v = *(const float4*)(lrow + i * 128 + lane * 4);
      s += __expf(v.x - m) + __expf(v.y - m) + __expf(v.z - m) + __expf(v.w - m);
    }
    if (lane < 26) {
      float4 v = *(const float4*)(lrow + 896 + lane * 4);
      s += __expf(v.x - m) + __expf(v.y - m) + __expf(v.z - m) + __expf(v.w - m);
    }
#pragma unroll
    for (int o = 16; o > 0; o >>= 1)
      s += __shfl_xor(s, o, 32);

    // ---- per-row scalar epilogue on lane 0 ----
    if (lane == 0) {
      const int t  = target[row];
      const float xt    = lrow[t];
      const float logpt = xt - m - __logf(s);
      const float pt    = __expf(logpt);
      // searchsorted(edges, pt, side='right') == count(edges <= pt)
      int bi = 0;
#pragma unroll
      for (int j = 0; j < NUM_EDGES; ++j)
        bi += (pt >= bin_edges[j]) ? 1 : 0;
      const float g    = bin_gammas[bi];
      const float base = 1.0f - pt + EPSF;
      acc += -__powf(base, g) * logpt;
    }
    buf ^= 1;
  }

  if (lane == 0) smem[16000 + warp] = acc;
  __syncthreads();
  if (tid == 0) {
    float p = 0.0f;
#pragma unroll
    for (int w = 0; w < WAVES_PER_BLK; ++w) p += smem[16000 + w];
    partials[blockIdx.x] = p;
  }
}

// Deterministic fixed-order reduction of per-block partials (graph replays
// must produce bit-identical sums; no float atomics anywhere).
__global__ __launch_bounds__(256) void focal_reduce_kernel(
    const float* __restrict__ partials, int n, float* __restrict__ out)
{
  __shared__ float sm[256];
  const int tid = threadIdx.x;
  float s = 0.0f;
  for (int i = tid; i < n; i += 256) s += partials[i];
  sm[tid] = s;
  __syncthreads();
#pragma unroll
  for (int o = 128; o > 0; o >>= 1) {
    if (tid < o) sm[tid] += sm[tid + o];
    __syncthreads();
  }
  if (tid == 0) out[0] = sm[0];
}

extern "C" void kernel_launch(void* const* d_in, const int* in_sizes, int n_in,
                              void* d_out, int out_size, void* d_ws, size_t ws_size,
                              hipStream_t stream) {
  const float* logits     = (const float*)d_in[0];
  const int*   target     = (const int*)  d_in[1];
  const float* bin_edges  = (const float*)d_in[2];
  const float* bin_gammas = (const float*)d_in[3];

  const int nrows = in_sizes[0] / C_CLASSES;   // 131072
  float* partials = (float*)d_ws;              // BLOCKS floats of scratch

  focal_main_kernel<<<BLOCKS, 256, 0, stream>>>(
      logits, target, bin_edges, bin_gammas, partials, nrows);
  focal_reduce_kernel<<<1, 256, 0, stream>>>(partials, BLOCKS, (float*)d_out);
}